// GraphSAGE_NET_60120952209414
// MI455X (gfx1250) — compile-verified
//
#include <hip/hip_runtime.h>

// GraphSAGE 2-layer forward for MI455X (gfx1250, wave32).
//
// Roofline: edge gather/scatter (2 x ~640MB of L2-resident traffic) dominates;
// GEMMs are ~3.3 GFLOP total and memory-bound, so we keep them in fp32 using
// the native V_WMMA_F32_16X16X4_F32 (no precision loss, no conversion cost).
// Scatter uses hardware global_atomic_add_f32 (no-return -> STOREcnt path).

typedef __attribute__((ext_vector_type(2))) float v2f;
typedef __attribute__((ext_vector_type(8))) float v8f;

#define D 64
#define WPAD 66  // padded LDS row stride (floats) -> conflict-free strided reads

// ---------------------------------------------------------------- zero scratch
__global__ void zero_f32(float* __restrict__ p, long n) {
  long i = (long)blockIdx.x * blockDim.x + threadIdx.x;
  long stride = (long)gridDim.x * blockDim.x;
  for (; i < n; i += stride) p[i] = 0.0f;
}

// ------------------------------------------------- edge scatter-add (mean num)
// 64 lanes per edge: one fp32 atomic per feature channel into agg[dst],
// lane 0 bumps the in-degree counter. All traffic stays in the 192MB L2.
__global__ void __launch_bounds__(256)
scatter_accum(const float* __restrict__ feat,
              const int* __restrict__ src,
              const int* __restrict__ dst,
              float* __restrict__ agg,
              float* __restrict__ cnt,
              int n_edges) {
  int t = blockIdx.x * blockDim.x + threadIdx.x;
  int e = t >> 6;
  int c = t & 63;
  if (e >= n_edges) return;
  int s = src[e];
  int d = dst[e];
  unsafeAtomicAdd(&agg[(size_t)d * D + c], feat[(size_t)s * D + c]);
  if (c == 0) unsafeAtomicAdd(&cnt[d], 1.0f);
}

// ------------------------------------------------------------- fused SAGE layer
// out[m, :] = act( mean[m,:] @ Wl^T + b + feat[m,:] @ Wr^T )
// One wave handles a 16-row tile x 64 output cols via 4 C-tiles (16x16 f32),
// accumulating K=64 per operand matrix with V_WMMA_F32_16X16X4_F32.
//
// fp32 WMMA fragment layouts (ISA 7.12.2):
//   A 16x4 : lanes 0-15 rows M=0..15 K={4k,4k+1}; lanes 16-31 same rows K={4k+2,4k+3}
//   B 4x16 : lanes 0-15 N=0..15 K={4k,4k+1};      lanes 16-31 same cols K={4k+2,4k+3}
//   C 16x16: VGPR r -> M = r + 8*(lane>=16), N = lane&15
__global__ void __launch_bounds__(256)
sage_layer(const float* __restrict__ feat,  // [N,64] input features
           const float* __restrict__ agg,   // [N,64] summed neighbor messages
           const float* __restrict__ cnt,   // [N]    in-degree counts
           const float* __restrict__ Wl,    // [64,64] row-major (n,k)
           const float* __restrict__ bias,  // [64]
           const float* __restrict__ Wr,    // [64,64] row-major (n,k)
           float* __restrict__ out,         // [N,64]
           int n_tiles, int n_nodes, int relu) {
  __shared__ float sW[2][D * WPAD];

  // Stage both weight matrices into LDS (bank-conflict-free padded rows).
  for (int j = threadIdx.x; j < D * D; j += blockDim.x) {
    int r = j >> 6, c = j & 63;
    sW[0][r * WPAD + c] = Wl[j];
    sW[1][r * WPAD + c] = Wr[j];
  }
  __syncthreads();

  int wave = threadIdx.x >> 5;
  int lane = threadIdx.x & 31;
  int tile = blockIdx.x * 8 + wave;  // 8 waves per 256-thread block
  if (tile >= n_tiles) return;

  int half = lane >> 4;   // 0: K pair {0,1}; 1: K pair {2,3}
  int lr   = lane & 15;
  int rowA = tile * 16 + lr;
  int rowC = rowA < n_nodes ? rowA : (n_nodes - 1);  // clamp tail loads

  float rc = 1.0f / fmaxf(cnt[rowC], 1.0f);
  const float* aggRow = agg  + (size_t)rowC * D;
  const float* xRow   = feat + (size_t)rowC * D;

  // Accumulators pre-loaded with the bias (same b[n] down every column).
  v8f c0, c1, c2, c3;
  {
    float bb0 = bias[ 0 + lr], bb1 = bias[16 + lr];
    float bb2 = bias[32 + lr], bb3 = bias[48 + lr];
#pragma unroll
    for (int r = 0; r < 8; ++r) { c0[r] = bb0; c1[r] = bb1; c2[r] = bb2; c3[r] = bb3; }
  }

  const float* wl = &sW[0][0];
  const float* wr = &sW[1][0];

#pragma unroll
  for (int kk = 0; kk < 16; ++kk) {
    int kb = kk * 4 + half * 2;  // this lane's K pair (8B aligned everywhere)

    // A fragments: mean (agg * 1/deg) and raw features.
    v2f am = *(const v2f*)(aggRow + kb);
    am *= rc;
    v2f ax = *(const v2f*)(xRow + kb);

    // B fragments from LDS: B[k][n] = W[n][k].
    v2f bl0 = *(const v2f*)(wl + ( 0 + lr) * WPAD + kb);
    v2f bl1 = *(const v2f*)(wl + (16 + lr) * WPAD + kb);
    v2f bl2 = *(const v2f*)(wl + (32 + lr) * WPAD + kb);
    v2f bl3 = *(const v2f*)(wl + (48 + lr) * WPAD + kb);
    v2f br0 = *(const v2f*)(wr + ( 0 + lr) * WPAD + kb);
    v2f br1 = *(const v2f*)(wr + (16 + lr) * WPAD + kb);
    v2f br2 = *(const v2f*)(wr + (32 + lr) * WPAD + kb);
    v2f br3 = *(const v2f*)(wr + (48 + lr) * WPAD + kb);

    c0 = __builtin_amdgcn_wmma_f32_16x16x4_f32(false, am, false, bl0, (short)0, c0, false, false);
    c1 = __builtin_amdgcn_wmma_f32_16x16x4_f32(false, am, false, bl1, (short)0, c1, false, false);
    c2 = __builtin_amdgcn_wmma_f32_16x16x4_f32(false, am, false, bl2, (short)0, c2, false, false);
    c3 = __builtin_amdgcn_wmma_f32_16x16x4_f32(false, am, false, bl3, (short)0, c3, false, false);
    c0 = __builtin_amdgcn_wmma_f32_16x16x4_f32(false, ax, false, br0, (short)0, c0, false, false);
    c1 = __builtin_amdgcn_wmma_f32_16x16x4_f32(false, ax, false, br1, (short)0, c1, false, false);
    c2 = __builtin_amdgcn_wmma_f32_16x16x4_f32(false, ax, false, br2, (short)0, c2, false, false);
    c3 = __builtin_amdgcn_wmma_f32_16x16x4_f32(false, ax, false, br3, (short)0, c3, false, false);
  }

  if (relu) {
#pragma unroll
    for (int r = 0; r < 8; ++r) {
      c0[r] = fmaxf(c0[r], 0.0f); c1[r] = fmaxf(c1[r], 0.0f);
      c2[r] = fmaxf(c2[r], 0.0f); c3[r] = fmaxf(c3[r], 0.0f);
    }
  }

  // Store: C layout M = r + 8*half, N = lr (+ column-tile base).
#pragma unroll
  for (int r = 0; r < 8; ++r) {
    int row = tile * 16 + half * 8 + r;
    if (row < n_nodes) {
      float* o = out + (size_t)row * D + lr;
      o[ 0] = c0[r];
      o[16] = c1[r];
      o[32] = c2[r];
      o[48] = c3[r];
    }
  }
}

// --------------------------------------------------------------------- launch
extern "C" void kernel_launch(void* const* d_in, const int* in_sizes, int n_in,
                              void* d_out, int out_size, void* d_ws, size_t ws_size,
                              hipStream_t stream) {
  const float* x   = (const float*)d_in[0];
  const int*   ei  = (const int*)d_in[1];
  const float* W1l = (const float*)d_in[2];
  const float* b1  = (const float*)d_in[3];
  const float* W1r = (const float*)d_in[4];
  const float* W2l = (const float*)d_in[5];
  const float* b2  = (const float*)d_in[6];
  const float* W2r = (const float*)d_in[7];
  float* out = (float*)d_out;

  long N = in_sizes[0] / D;      // 100000
  int  E = in_sizes[1] / 2;      // 1250000
  const int* src = ei;           // edge_index[0]
  const int* dst = ei + E;       // edge_index[1]

  // Scratch layout: agg [N*64] | cnt [N] | h [N*64]  (~52 MB for N=100k)
  float* agg = (float*)d_ws;
  float* cnt = agg + (size_t)N * D;
  float* h   = cnt + N;

  int n_tiles = (int)((N + 15) / 16);
  int lblocks = (n_tiles + 7) / 8;
  int sblocks = (E + 3) / 4;       // 4 edges per 256-thread block
  long zn = N * (long)(D + 1);     // agg + cnt are contiguous

  // Layer 1: h = relu(mean(x) @ W1l^T + b1 + x @ W1r^T)
  zero_f32<<<1024, 256, 0, stream>>>(agg, zn);
  scatter_accum<<<sblocks, 256, 0, stream>>>(x, src, dst, agg, cnt, E);
  sage_layer<<<lblocks, 256, 0, stream>>>(x, agg, cnt, W1l, b1, W1r, h,
                                          n_tiles, (int)N, 1);

  // Layer 2: out = mean(h) @ W2l^T + b2 + h @ W2r^T
  zero_f32<<<1024, 256, 0, stream>>>(agg, zn);
  scatter_accum<<<sblocks, 256, 0, stream>>>(h, src, dst, agg, cnt, E);
  sage_layer<<<lblocks, 256, 0, stream>>>(h, agg, cnt, W2l, b2, W2r, out,
                                          n_tiles, (int)N, 0);
}